// BoardEncoder_40407052321081
// MI455X (gfx1250) — compile-verified
//
#include <hip/hip_runtime.h>

typedef __attribute__((ext_vector_type(2))) float v2f;
typedef __attribute__((ext_vector_type(4))) float v4f;
typedef __attribute__((ext_vector_type(8))) float v8f;

#define B_TOTAL 16384
#define NSEG    54
#define DIN     64
#define DHID    4
#define DOUT    256
#define KTOT    (NSEG * DHID)   // 216
#define TILE_B  16
#define HSTRIDE 220             // padded LDS row stride (floats) to spread banks

// ---------------------------------------------------------------------------
// Prep: transpose Wf [256][216] -> WfT [216][256] so the WMMA B fragments load
// as contiguous 64B rows per half-wave. Tiny (55K elements), runs once/launch.
// ---------------------------------------------------------------------------
__global__ __launch_bounds__(256)
void wf_transpose_kernel(const float* __restrict__ Wf, float* __restrict__ WfT) {
    int idx = blockIdx.x * 256 + threadIdx.x;
    if (idx < DOUT * KTOT) {
        int o = idx / KTOT;
        int k = idx - o * KTOT;
        WfT[k * DOUT + o] = Wf[idx];
    }
}

// ---------------------------------------------------------------------------
// Fused hexconv + ReLU + [16,216]x[216,256] GEMM per workgroup of 16 batches.
// ---------------------------------------------------------------------------
__global__ __launch_bounds__(256)
void hexconv_gemm_kernel(const float* __restrict__ x,
                         const float* __restrict__ nbr,
                         const float* __restrict__ Wc,
                         const float* __restrict__ bc,
                         const float* __restrict__ Wd,
                         const float* __restrict__ bd,
                         const float* __restrict__ WfT,
                         const float* __restrict__ bf,
                         float* __restrict__ out) {
    __shared__ float hLDS[TILE_B][HSTRIDE];

    const int tid   = threadIdx.x;
    const int lane  = tid & 31;
    const int wave  = tid >> 5;     // 0..7
    const int l16   = lane & 15;
    const int khalf = lane >> 4;    // 0 or 1 (half-wave)
    const int bbase = blockIdx.x * TILE_B;

    // ---- preload per-lane weight fragments into registers ----
    // x path: lane covers d = 2*lane, 2*lane+1
    v2f wc[DHID];
#pragma unroll
    for (int h = 0; h < DHID; ++h)
        wc[h] = *(const v2f*)(Wc + h * DIN + 2 * lane);

    // neighbor path: flattened 384-float row; lane's float4 #i covers
    // element idx = 128*i + 4*lane  ->  direction k = 2*i + khalf, d = 4*l16
    v4f wd[3][DHID];
#pragma unroll
    for (int i = 0; i < 3; ++i) {
        int k = 2 * i + khalf;
#pragma unroll
        for (int h = 0; h < DHID; ++h)
            wd[i][h] = *(const v4f*)(Wd + (k * DHID + h) * DIN + 4 * l16);
    }

    float bsum[DHID];
#pragma unroll
    for (int h = 0; h < DHID; ++h) {
        float s = bc[h];
#pragma unroll
        for (int k = 0; k < 6; ++k) s += bd[k * DHID + h];
        bsum[h] = s;
    }

    // ---- stage 1: one wave per (b,n) pair, 108 pairs per wave ----
    for (int p = wave; p < TILE_B * NSEG; p += 8) {
        int bl = p / NSEG;
        int n  = p - bl * NSEG;
        const float* xr = x   + ((size_t)(bbase + bl) * NSEG + n) * DIN;
        const float* nr = nbr + ((size_t)(bbase + bl) * NSEG + n) * (6 * DIN);

        v2f xv = *(const v2f*)(xr + 2 * lane);                 // b64 coalesced
        float acc[DHID];
#pragma unroll
        for (int h = 0; h < DHID; ++h)
            acc[h] = xv.x * wc[h].x + xv.y * wc[h].y;

#pragma unroll
        for (int i = 0; i < 3; ++i) {
            v4f nv = *(const v4f*)(nr + 128 * i + 4 * lane);   // b128 coalesced
#pragma unroll
            for (int h = 0; h < DHID; ++h)
                acc[h] += nv.x * wd[i][h].x + nv.y * wd[i][h].y +
                          nv.z * wd[i][h].z + nv.w * wd[i][h].w;
        }

        // wave32 butterfly reduction of the 4 hidden sums
#pragma unroll
        for (int m = 16; m >= 1; m >>= 1)
#pragma unroll
            for (int h = 0; h < DHID; ++h)
                acc[h] += __shfl_xor(acc[h], m, 32);

        if (lane < DHID) {
            float v = (lane == 0) ? acc[0] : (lane == 1) ? acc[1]
                    : (lane == 2) ? acc[2] : acc[3];
            v += (lane == 0) ? bsum[0] : (lane == 1) ? bsum[1]
               : (lane == 2) ? bsum[2] : bsum[3];
            hLDS[bl][n * DHID + lane] = v > 0.0f ? v : 0.0f;   // ReLU
        }
    }

    __syncthreads();

    // ---- stage 2: WMMA GEMM  out[16, 256] = h[16,216] @ WfT[216,256] + bf ----
    // Each wave owns two 16-wide N-tiles: columns [wave*32, wave*32+32).
    const int ob0 = wave * 32;
    const int ob1 = ob0 + 16;

    v8f c0 = {};
    v8f c1 = {};

    for (int k = 0; k < KTOT; k += 4) {
        // A fragment (16x4 f32): VGPR0 = K=0 (lanes 0-15) / K=2 (lanes 16-31),
        // VGPR1 = K=1 / K=3  ->  lane reads float2 at column k + 2*khalf.
        v2f a = *(const v2f*)(&hLDS[l16][k + 2 * khalf]);

        // B fragments (4x16 f32), symmetric layout: V0 = row k (lo) / k+2 (hi),
        // V1 = row k+1 (lo) / k+3 (hi). WfT rows are contiguous -> 64B/halfwave.
        const float* wrow = WfT + (size_t)(k + 2 * khalf) * DOUT;
        v2f b0v, b1v;
        b0v.x = wrow[ob0 + l16];
        b0v.y = wrow[DOUT + ob0 + l16];
        b1v.x = wrow[ob1 + l16];
        b1v.y = wrow[DOUT + ob1 + l16];

        c0 = __builtin_amdgcn_wmma_f32_16x16x4_f32(false, a, false, b0v,
                                                   (short)0, c0, false, false);
        c1 = __builtin_amdgcn_wmma_f32_16x16x4_f32(false, a, false, b1v,
                                                   (short)0, c1, false, false);
    }

    // ---- epilogue: D layout VGPR j -> M=j (lanes 0-15) / M=8+j (lanes 16-31)
    float bias0 = bf[ob0 + l16];
    float bias1 = bf[ob1 + l16];
    const int mbase = bbase + khalf * 8;
#pragma unroll
    for (int j = 0; j < 8; ++j) {
        out[(size_t)(mbase + j) * DOUT + ob0 + l16] = c0[j] + bias0;
        out[(size_t)(mbase + j) * DOUT + ob1 + l16] = c1[j] + bias1;
    }
}

// ---------------------------------------------------------------------------
extern "C" void kernel_launch(void* const* d_in, const int* in_sizes, int n_in,
                              void* d_out, int out_size, void* d_ws, size_t ws_size,
                              hipStream_t stream) {
    const float* x    = (const float*)d_in[0];
    const float* nbrs = (const float*)d_in[1];
    const float* Wc   = (const float*)d_in[2];
    const float* bc   = (const float*)d_in[3];
    const float* Wd   = (const float*)d_in[4];
    const float* bd   = (const float*)d_in[5];
    const float* Wf   = (const float*)d_in[6];
    const float* bf   = (const float*)d_in[7];
    float* out = (float*)d_out;
    float* WfT = (float*)d_ws;   // 216*256*4 = 221184 bytes of scratch

    wf_transpose_kernel<<<(DOUT * KTOT + 255) / 256, 256, 0, stream>>>(Wf, WfT);

    hexconv_gemm_kernel<<<B_TOTAL / TILE_B, 256, 0, stream>>>(
        x, nbrs, Wc, bc, Wd, bd, WfT, bf, out);
}